// LoTDForestEncoding_21242908246572
// MI455X (gfx1250) — compile-verified
//
#include <hip/hip_runtime.h>

// LoTD forest hash-grid encoding for MI455X (gfx1250).
// Gather-bound kernel: 134 MB param table is L2-resident (192 MB L2);
// 2 GB of random 8B gathers + 256 MB streaming stores.
// gfx1250 paths: async gather-to-LDS (ASYNCcnt) software pipeline,
// ds_load consumption from LDS, non-temporal output stores to keep the
// hash tables resident in L2.

#define LL 16
#define TT 131072
#define TMASK (TT - 1)

__constant__ int c_res[LL] = {16, 22, 30, 41, 55, 75, 102, 139,
                              188, 256, 348, 472, 642, 872, 1184, 1608};

#define AS1 __attribute__((address_space(1)))
#define AS3 __attribute__((address_space(3)))

typedef int v2i32 __attribute__((__vector_size__(2 * sizeof(int))));
typedef float v2f __attribute__((ext_vector_type(2)));

#if defined(__HIP_DEVICE_COMPILE__) && __has_builtin(__builtin_amdgcn_global_load_async_to_lds_b64)
#define HAVE_ASYNC_GATHER 1
#else
#define HAVE_ASYNC_GATHER 0
#endif

struct Geom {
    int i0[3], i1[3];
    float w[3];
};

__device__ __forceinline__ Geom make_geom(int R, float x01, float y01, float z01) {
    Geom g;
    const float rm1 = (float)(R - 1);
    float p[3] = {x01 * rm1, y01 * rm1, z01 * rm1};
#pragma unroll
    for (int d = 0; d < 3; ++d) {
        float f = floorf(p[d]);
        g.w[d] = p[d] - f;
        int i = (int)f;
        int lo = i < 0 ? 0 : i;
        g.i0[d] = lo > R - 1 ? R - 1 : lo;
        int hi = i + 1;
        hi = hi < 0 ? 0 : hi;
        g.i1[d] = hi > R - 1 ? R - 1 : hi;
    }
    return g;
}

__device__ __forceinline__ unsigned hash3(unsigned cx, unsigned cy, unsigned cz) {
    return (cx ^ (cy * 2654435761u) ^ (cz * 805459861u)) & TMASK;
}

__global__ __launch_bounds__(256) void lotd_kernel(
    const float* __restrict__ bx, const float* __restrict__ params,
    const int* __restrict__ inds, float* __restrict__ out, int npts) {
    const int tid = threadIdx.x;
    const int n = blockIdx.x * 256 + tid;
    const int nn = n < npts ? n : (npts - 1);
    const bool active = (n < npts);

    const float x01 = bx[3 * (size_t)nn + 0] * 0.5f + 0.5f;
    const float y01 = bx[3 * (size_t)nn + 1] * 0.5f + 0.5f;
    const float z01 = bx[3 * (size_t)nn + 2] * 0.5f + 0.5f;
    const int tree = inds[nn];
    const float2* tb = (const float2*)params + (size_t)tree * ((size_t)LL * TT);
    v2f* ob = (v2f*)out + (size_t)nn * LL;

#if HAVE_ASYNC_GATHER
    // Double-buffered async gather-to-LDS pipeline: issue level l+1's 8 corner
    // gathers, then s_wait_asynccnt 8 to guarantee level l's (older, in-order)
    // gathers have landed in LDS. Per-lane private LDS slots -> no barriers.
    __shared__ float2 stage[2][256 * 8];
    // Explicit AS3 pointers so reads lower to ds_load (not flat_load).
    AS3 v2i32* slot0w = (AS3 v2i32*)(void*)&stage[0][tid * 8];
    AS3 v2i32* slot1w = (AS3 v2i32*)(void*)&stage[1][tid * 8];
    const AS3 float* slot0r = (const AS3 float*)(void*)&stage[0][tid * 8];
    const AS3 float* slot1r = (const AS3 float*)(void*)&stage[1][tid * 8];

    auto issue = [&](int l, const Geom& g) {
        const float2* lvl = tb + (size_t)l * TT;
        AS3 v2i32* s3 = (l & 1) ? slot1w : slot0w;
#pragma unroll
        for (int c = 0; c < 8; ++c) {
            const int bxm = (c >> 2) & 1, bym = (c >> 1) & 1, bzm = c & 1;
            unsigned idx = hash3(bxm ? g.i1[0] : g.i0[0],
                                 bym ? g.i1[1] : g.i0[1],
                                 bzm ? g.i1[2] : g.i0[2]);
            __builtin_amdgcn_global_load_async_to_lds_b64(
                (AS1 v2i32*)(void*)(lvl + idx), s3 + c, 0, 0);
        }
    };

    Geom g0 = make_geom(c_res[0], x01, y01, z01);
    issue(0, g0);
    float wx = g0.w[0], wy = g0.w[1], wz = g0.w[2];

#pragma unroll 1
    for (int l = 0; l < LL; ++l) {
        float wxn = wx, wyn = wy, wzn = wz;
        if (l + 1 < LL) {
            Geom gn = make_geom(c_res[l + 1], x01, y01, z01);
            issue(l + 1, gn);
            wxn = gn.w[0]; wyn = gn.w[1]; wzn = gn.w[2];
            asm volatile("s_wait_asynccnt 0x8" ::: "memory");
        } else {
            asm volatile("s_wait_asynccnt 0x0" ::: "memory");
        }
        const AS3 float* sf = (l & 1) ? slot1r : slot0r;
        float ax = 0.f, ay = 0.f;
#pragma unroll
        for (int c = 0; c < 8; ++c) {
            const int bxm = (c >> 2) & 1, bym = (c >> 1) & 1, bzm = c & 1;
            float wc = (bxm ? wx : 1.f - wx) * (bym ? wy : 1.f - wy) *
                       (bzm ? wz : 1.f - wz);
            float fx = sf[2 * c + 0];
            float fy = sf[2 * c + 1];
            ax = fmaf(wc, fx, ax);
            ay = fmaf(wc, fy, ay);
        }
        if (active) {
            v2f o;
            o.x = ax;
            o.y = ay;
            __builtin_nontemporal_store(o, ob + l);  // keep L2 for the tables
        }
        wx = wxn; wy = wyn; wz = wzn;
    }
#else
    // Fallback: direct global_load_b64 gathers (8 in flight per level).
#pragma unroll 1
    for (int l = 0; l < LL; ++l) {
        Geom g = make_geom(c_res[l], x01, y01, z01);
        const float2* lvl = tb + (size_t)l * TT;
        float ax = 0.f, ay = 0.f;
#pragma unroll
        for (int c = 0; c < 8; ++c) {
            const int bxm = (c >> 2) & 1, bym = (c >> 1) & 1, bzm = c & 1;
            unsigned idx = hash3(bxm ? g.i1[0] : g.i0[0],
                                 bym ? g.i1[1] : g.i0[1],
                                 bzm ? g.i1[2] : g.i0[2]);
            float wc = (bxm ? g.w[0] : 1.f - g.w[0]) *
                       (bym ? g.w[1] : 1.f - g.w[1]) *
                       (bzm ? g.w[2] : 1.f - g.w[2]);
            float2 f = lvl[idx];
            ax = fmaf(wc, f.x, ax);
            ay = fmaf(wc, f.y, ay);
        }
        if (active) {
            v2f o;
            o.x = ax;
            o.y = ay;
            __builtin_nontemporal_store(o, ob + l);
        }
    }
#endif
}

extern "C" void kernel_launch(void* const* d_in, const int* in_sizes, int n_in,
                              void* d_out, int out_size, void* d_ws, size_t ws_size,
                              hipStream_t stream) {
    (void)n_in; (void)out_size; (void)d_ws; (void)ws_size;
    const float* bx = (const float*)d_in[0];      // block_x: N x 3 f32
    const float* params = (const float*)d_in[1];  // (8, 16, 131072, 2) f32
    const int* inds = (const int*)d_in[2];        // block_inds: N i32
    float* out = (float*)d_out;                   // N x 32 f32
    const int npts = in_sizes[2];
    const int blocks = (npts + 255) / 256;
    lotd_kernel<<<blocks, 256, 0, stream>>>(bx, params, inds, out, npts);
}